// GraphEncoder_23802708754725
// MI455X (gfx1250) — compile-verified
//
#include <hip/hip_runtime.h>

// GINEConv ×3 for MI455X (gfx1250, wave32).
// All GEMMs run on V_WMMA_F32_16X16X4_F32 (exact fp32, matches reference).
// Kernel is HBM/atomic bound (~2.6 GB traffic, ~115us @ 23.3 TB/s), so fp32
// matrix path is free; WMMA just offloads the dense math from VALU.

#define D   64
#define ED  32
#define OD  (4 * D)      // d_out row stride (concat of 4 blocks)
#define TSTRIDE 68       // padded LDS tile stride (bank-conflict free)

typedef float v2f __attribute__((ext_vector_type(2)));
typedef float v8f __attribute__((ext_vector_type(8)));

static __device__ __forceinline__ v8f wmma4(v2f a, v2f b, v8f c) {
  // D(16x16,f32) += A(16x4,f32) * B(4x16,f32)
  return __builtin_amdgcn_wmma_f32_16x16x4_f32(
      /*neg_a=*/false, a, /*neg_b=*/false, b,
      /*c_mod=*/(short)0, c, /*reuse_a=*/false, /*reuse_b=*/false);
}

__global__ void fill_zero_kernel(float* __restrict__ p, int n) {
  int i = blockIdx.x * blockDim.x + threadIdx.x;
  if (i < n) p[i] = 0.0f;
}

// x[N][64] -> d_out[:, 0:64] (row stride 256)
__global__ void copy_x_kernel(const float* __restrict__ x,
                              float* __restrict__ out, int N) {
  int i = blockIdx.x * blockDim.x + threadIdx.x;
  if (i < N * D) {
    int n = i >> 6, c = i & 63;
    out[n * OD + c] = x[i];
  }
}

// One wave per 16-edge tile:
//   e   = edge_attr[16x32] @ Wle[32x64] + ble        (32 fp32 WMMAs)
//   m   = relu(x[src] + e)
//   aggr[dst] += m                                   (global_atomic_add_f32)
__global__ __launch_bounds__(256) void edge_msg_kernel(
    const float* __restrict__ xprev,   // d_out + l*D, row stride OD
    const int*   __restrict__ src,
    const int*   __restrict__ dst,
    const float* __restrict__ eattr,   // [E][ED]
    const float* __restrict__ Wle,     // [ED][D]
    const float* __restrict__ ble,     // [D]
    float*       __restrict__ aggr,    // [N][D]
    int E) {
  const int lane = threadIdx.x & 31;
  const int half = lane >> 4;          // 0 | 1 (K sub-pair select)
  const int mrow = lane & 15;
  const int wid  = blockIdx.x * (blockDim.x >> 5) + (threadIdx.x >> 5);
  const int base = wid * 16;
  if (base >= E) return;               // wave-uniform exit

  // ---- A: 16 edges x 32 attrs, fp32 WMMA A layout (K pair per lane-half)
  int ea = base + mrow; if (ea >= E) ea = E - 1;   // clamp (loads only)
  const float* arow = eattr + ea * ED;
  v2f A[8];
#pragma unroll
  for (int kk = 0; kk < 8; ++kk)
    A[kk] = *(const v2f*)(arow + kk * 4 + 2 * half);

  const v8f vzero = {0.f, 0.f, 0.f, 0.f, 0.f, 0.f, 0.f, 0.f};
  v8f acc[4];

#pragma unroll
  for (int t = 0; t < 4; ++t) {
    const int col = t * 16 + mrow;     // B/N lane mapping
    acc[t] = vzero;
#pragma unroll
    for (int kk = 0; kk < 8; ++kk) {
      const int k = kk * 4 + 2 * half;
      v2f b;
      b.x = Wle[k * D + col];
      b.y = Wle[(k + 1) * D + col];
      acc[t] = wmma4(A[kk], b, acc[t]);
    }
  }

  // ---- rows this lane owns in the C/D layout: M = v + 8*half
  int  sv[8], dv[8];
  bool ok[8];
#pragma unroll
  for (int v = 0; v < 8; ++v) {
    int e  = base + v + 8 * half;
    ok[v]  = (e < E);
    int ec = ok[v] ? e : (E - 1);
    sv[v]  = src[ec];
    dv[v]  = dst[ec];
  }

#pragma unroll
  for (int t = 0; t < 4; ++t) {
    const int   col = t * 16 + mrow;
    const float bc  = ble[col];
#pragma unroll
    for (int v = 0; v < 8; ++v) {
      float lin = acc[t][v] + bc;
      float m   = xprev[sv[v] * OD + col] + lin;
      m = m > 0.f ? m : 0.f;
      if (ok[v]) unsafeAtomicAdd(&aggr[dv[v] * D + col], m);
    }
  }
}

// One wave per 16-node tile:
//   xh = x + aggr ; T = relu(xh@W1+b1) ; h = T@W2+b2
//   out = alpha*x + (1-alpha)*h    (written to d_out block l+1)
// D-layout -> A-layout re-swizzle between the two GEMMs goes through LDS.
__global__ __launch_bounds__(256) void node_mlp_kernel(
    const float* __restrict__ xprev,   // d_out + l*D, row stride OD
    const float* __restrict__ aggr,    // [N][D]
    const float* __restrict__ W1, const float* __restrict__ b1,
    const float* __restrict__ W2, const float* __restrict__ b2,
    const float* __restrict__ alpha, int l,
    float* __restrict__ outnext,       // d_out + (l+1)*D, row stride OD
    int N) {
  __shared__ float lds[8 * 16 * TSTRIDE];   // 8 waves/block, 16x64(+pad) each

  const int lane  = threadIdx.x & 31;
  const int half  = lane >> 4;
  const int mrow  = lane & 15;
  const int wslot = threadIdx.x >> 5;
  const int base  = (blockIdx.x * (blockDim.x >> 5) + wslot) * 16;
  float* tile = lds + wslot * (16 * TSTRIDE);
  const float al = alpha[l];

  // ---- A1 = (x + aggr) rows, WMMA A layout
  int na = base + mrow; if (na >= N) na = N - 1;
  v2f A[16];
#pragma unroll
  for (int kk = 0; kk < 16; ++kk) {
    const int k = kk * 4 + 2 * half;
    v2f xp = *(const v2f*)(xprev + na * OD + k);
    v2f ag = *(const v2f*)(aggr  + na * D  + k);
    A[kk].x = xp.x + ag.x;
    A[kk].y = xp.y + ag.y;
  }

  const v8f vzero = {0.f, 0.f, 0.f, 0.f, 0.f, 0.f, 0.f, 0.f};

  // ---- GEMM1 + bias + relu -> LDS tile (row-major 16x64, stride 68)
#pragma unroll
  for (int t = 0; t < 4; ++t) {
    const int col = t * 16 + mrow;
    v8f acc = vzero;
#pragma unroll
    for (int kk = 0; kk < 16; ++kk) {
      const int k = kk * 4 + 2 * half;
      v2f b;
      b.x = W1[k * D + col];
      b.y = W1[(k + 1) * D + col];
      acc = wmma4(A[kk], b, acc);
    }
    const float bb = b1[col];
#pragma unroll
    for (int v = 0; v < 8; ++v) {
      float val = acc[v] + bb;
      val = val > 0.f ? val : 0.f;
      tile[(v + 8 * half) * TSTRIDE + col] = val;   // D-layout scatter
    }
  }

  __syncthreads();   // cross-lane LDS visibility (ds store -> ds load)

  // ---- A2 = T rows from LDS, WMMA A layout (conflict-free: stride 68)
  v2f A2[16];
#pragma unroll
  for (int kk = 0; kk < 16; ++kk) {
    const int k = kk * 4 + 2 * half;
    A2[kk] = *(const v2f*)(tile + mrow * TSTRIDE + k);
  }

  // ---- GEMM2 + bias + skip connection -> d_out block l+1
#pragma unroll
  for (int t = 0; t < 4; ++t) {
    const int col = t * 16 + mrow;
    v8f acc = vzero;
#pragma unroll
    for (int kk = 0; kk < 16; ++kk) {
      const int k = kk * 4 + 2 * half;
      v2f b;
      b.x = W2[k * D + col];
      b.y = W2[(k + 1) * D + col];
      acc = wmma4(A2[kk], b, acc);
    }
    const float bb = b2[col];
#pragma unroll
    for (int v = 0; v < 8; ++v) {
      const int node = base + v + 8 * half;
      if (node < N) {
        float h  = acc[v] + bb;
        float xp = xprev[node * OD + col];
        outnext[node * OD + col] = al * xp + (1.f - al) * h;
      }
    }
  }
}

extern "C" void kernel_launch(void* const* d_in, const int* in_sizes, int n_in,
                              void* d_out, int out_size, void* d_ws, size_t ws_size,
                              hipStream_t stream) {
  const float* x     = (const float*)d_in[0];
  const int*   eidx  = (const int*)  d_in[1];
  const float* eattr = (const float*)d_in[2];
  const float* Wle   = (const float*)d_in[3];
  const float* ble   = (const float*)d_in[4];
  const float* W1    = (const float*)d_in[5];
  const float* b1    = (const float*)d_in[6];
  const float* W2    = (const float*)d_in[7];
  const float* b2    = (const float*)d_in[8];
  const float* alpha = (const float*)d_in[9];
  float* out = (float*)d_out;

  const int N = in_sizes[0] / D;     // 100000
  const int E = in_sizes[2] / ED;    // 1250000
  const int* src = eidx;             // edge_index[0]
  const int* dst = eidx + E;         // edge_index[1]

  float* aggr = (float*)d_ws;        // N*D floats of scratch

  // xs[0] = x into d_out block 0
  copy_x_kernel<<<(N * D + 255) / 256, 256, 0, stream>>>(x, out, N);

  const int etiles  = (E + 15) / 16;
  const int eblocks = (etiles + 7) / 8;    // 8 waves / block
  const int ntiles  = (N + 15) / 16;
  const int nblocks = (ntiles + 7) / 8;

  for (int l = 0; l < 3; ++l) {
    fill_zero_kernel<<<(N * D + 255) / 256, 256, 0, stream>>>(aggr, N * D);
    edge_msg_kernel<<<eblocks, 256, 0, stream>>>(
        out + l * D, src, dst, eattr,
        Wle + l * ED * D, ble + l * D, aggr, E);
    node_mlp_kernel<<<nblocks, 256, 0, stream>>>(
        out + l * D, aggr,
        W1 + l * D * D, b1 + l * D, W2 + l * D * D, b2 + l * D,
        alpha, l, out + (l + 1) * D, N);
  }
}